// timeModule_25434796327365
// MI455X (gfx1250) — compile-verified
//
#include <hip/hip_runtime.h>

// MI455X (gfx1250) LSTM-gate kernel, round 4.
// Roofline: ~201 MB traffic @ 23.3 TB/s => ~8.6us floor; compute ~1 GFLOP.
// Round-3 was ~7.4 B moved per issued instruction (issue-bound tail): scalar
// b32 global ops + per-tile exec masking for the odd channel store.
// Round-4 stages each batch row through wave-private LDS (same-wave DS ops
// are in-order, no barriers): global traffic becomes pure 512B b128 bursts,
// B operands come from one ds_load_2addr per tile, results leave via one
// unconditional ds_store_2addr per tile (upper-half dead value -> dump row),
// eliminating all divergence. Bank-padded row stride 140 floats keeps the
// two lane-halves on disjoint banks and offsets within ds_*2addr immediates.

typedef float v2f __attribute__((ext_vector_type(2)));
typedef float v4f __attribute__((ext_vector_type(4)));
typedef float v8f __attribute__((ext_vector_type(8)));

#if __has_builtin(__builtin_amdgcn_tanhf)
__device__ __forceinline__ float fast_tanh(float x) {
    return __builtin_amdgcn_tanhf(x);            // V_TANH_F32: 1 instruction
}
#else
__device__ __forceinline__ float fast_tanh(float x) {
    return __builtin_fmaf(2.0f, __builtin_amdgcn_rcpf(1.0f + __expf(-(x + x))), -1.0f);
}
#endif

__device__ __forceinline__ float fast_sig(float x) {
    // sigmoid(x) = 0.5*tanh(x/2) + 0.5
    return __builtin_fmaf(0.5f, fast_tanh(0.5f * x), 0.5f);
}

// One LSTM cell output from its gate pre-activations (f*c0 == 0):
// h = sigmoid(o) * tanh( sigmoid(i) * tanh(g) )
__device__ __forceinline__ float cell(float gi, float gg, float go) {
    const float i = fast_sig(gi);
    const float g = fast_tanh(gg);
    const float o = fast_sig(go);
    return o * fast_tanh(i * g);
}

// D-row -> original PyTorch gate-row permutation.
// Rows {0,1,2}=(i0,g0,o0), {3,4,5}=(i1,g1,o1) live in lower lane-half
// (VGPR 0..5); rows {8,9,10}=(i2,g2,o2) in upper lane-half (VGPR 0..2).
__device__ __forceinline__ int row_gate(int m) {
    const int tbl[16] = {0, 6, 9, 1, 7, 10, -1, -1, 2, 8, 11, -1, -1, -1, -1, -1};
    return tbl[m & 15];
}

// ws layout (floats): [0..63]  A-matrix in per-lane order (lane l -> ws[2l..])
//                     [64..79] permuted bias row vector (index = D row m)
__global__ void lstm_setup(const float* __restrict__ W,
                           const float* __restrict__ b_ih,
                           const float* __restrict__ b_hh,
                           float* __restrict__ ws) {
    const int  lane = threadIdx.x & 31;
    const int  n    = lane & 15;
    const bool hi   = lane >= 16;
    const int  g    = row_gate(n);
    // A: 16x4 fp32, lanes 0-15 hold M=lane K={0,1}, lanes 16-31 K={2,3}.
    ws[2 * lane]     = (g >= 0) ? W[g * 3 + (hi ? 2 : 0)] : 0.0f;
    ws[2 * lane + 1] = (g >= 0 && !hi) ? W[g * 3 + 1] : 0.0f;
    if (!hi) ws[64 + n] = (g >= 0) ? (b_ih[g] + b_hh[g]) : 0.0f;
}

// LDS layout per wave (floats, row stride S=140 for banks + 2addr range):
//   in:  k0@0, k1@140, k2@280, zeros(K=3)@420   (560 floats)
//   out: ch0@560, ch1@700, ch2@840, dump@980    (560 floats)
#define S    140
#define WSZ  1120

__global__ __launch_bounds__(256) void lstm_gates_wmma(
    const float* __restrict__ x,      // (B, 3, 128)
    const float* __restrict__ ws,     // precomputed A + bias (L2-hot)
    float* __restrict__ out)          // (B, 3, 128)
{
    __shared__ float L[8 * WSZ];      // 35,840 B per block (8 wave32)

    const int  lane = threadIdx.x & 31;
    const int  wave = threadIdx.x >> 5;
    const int  bidx = (blockIdx.x * 8 + wave) * 2;  // 2 batch rows per wave
    const int  n    = lane & 15;                    // column (t within tile)
    const bool hi   = lane >= 16;                   // upper lane-half

    // WMMA constants: one b64 + one 32B load, L2-resident.
    const v2f a = *(const v2f*)(ws + 2 * lane);
    const v8f c = *(const v8f*)(ws + 64 + (hi ? 8 : 0));

    const int inB  = wave * WSZ;
    const int outB = inB + 4 * S;
    // Zero the K=3 row once: upper lanes' second B element reads it, and it
    // must be finite 0.0 (WMMA propagates NaN even through a zero A column).
    *(v4f*)(&L[inB + 3 * S + 4 * lane]) = (v4f){0.f, 0.f, 0.f, 0.f};

    const int rb = inB  + (hi ? 2 * S : 0) + n;  // K0 row: k=0 (lo) / k=2 (hi)
    const int wb = outB + (hi ? 2 * S : 0) + n;  // ch0 (lo) / ch2 (hi)

#pragma unroll
    for (int r = 0; r < 2; ++r) {
        const float* xb = x   + (size_t)(bidx + r) * 384;
        float*       ob = out + (size_t)(bidx + r) * 384;

        // Stage input row: 3 contiguous 512B b128 loads -> padded LDS rows.
#pragma unroll
        for (int it = 0; it < 3; ++it)
            *(v4f*)(&L[inB + it * S + 4 * lane]) =
                *(const v4f*)(xb + it * 128 + 4 * lane);

#pragma unroll
        for (int t0 = 0; t0 < 128; t0 += 16) {
            // B matrix 4x16: one ds_load_2addr (offsets t0, t0+S).
            v2f bm;
            bm.x = L[rb + t0];        // K0: k=0 (lo) / k=2 (hi)
            bm.y = L[rb + t0 + S];    // K1: k=1 (lo) / zeros row (hi)

            const v8f d = __builtin_amdgcn_wmma_f32_16x16x4_f32(
                false, a, false, bm, (short)0, c, false, false);

            const float hA = cell(d[0], d[1], d[2]);  // ch0 (lo) / ch2 (hi)
            const float hB = cell(d[3], d[4], d[5]);  // ch1 (lo) / dead (hi)

            // One unconditional ds_store_2addr: no exec masking anywhere.
            L[wb + t0]     = hA;      // ch0 (lo) / ch2 (hi)
            L[wb + t0 + S] = hB;      // ch1 (lo) / dump row (hi)
        }

        // Drain staged outputs: 3 contiguous 512B b128 stores.
#pragma unroll
        for (int it = 0; it < 3; ++it)
            *(v4f*)(ob + it * 128 + 4 * lane) =
                *(const v4f*)(&L[outB + it * S + 4 * lane]);
    }
}

extern "C" void kernel_launch(void* const* d_in, const int* in_sizes, int n_in,
                              void* d_out, int out_size, void* d_ws, size_t ws_size,
                              hipStream_t stream) {
    (void)in_sizes; (void)n_in; (void)ws_size; (void)out_size;
    const float* x    = (const float*)d_in[0];   // (65536, 3, 128)
    const float* W_ih = (const float*)d_in[1];   // (12, 3)
    // d_in[2] = W_hh: dead (h0 = c0 = 0); only b_hh contributes.
    const float* b_ih = (const float*)d_in[3];   // (12,)
    const float* b_hh = (const float*)d_in[4];   // (12,)
    float* out = (float*)d_out;                  // (65536, 3, 128)
    float* ws  = (float*)d_ws;                   // 320 B used

    // 1) Materialize permuted A-matrix + bias into workspace (one wave).
    lstm_setup<<<1, 32, 0, stream>>>(W_ih, b_ih, b_hh, ws);

    // 2) Stream kernel: 8 wave32 per block, 2 batch rows per wave.
    const int B = 65536;
    dim3 block(256);
    dim3 grid(B / 16);               // 4096 blocks
    lstm_gates_wmma<<<grid, block, 0, stream>>>(x, ws, out);
}